// CMB_H_OMBH2_76613626626430
// MI455X (gfx1250) — compile-verified
//
#include <hip/hip_runtime.h>
#include <hip/hip_bf16.h>

typedef __attribute__((ext_vector_type(2))) float v2f;
typedef __attribute__((ext_vector_type(8))) float v8f;

#define HID_PAD 112   // 100 padded to 7*16
#define NCH 256       // channels (columns of y)
#define NKNOT 128
#define NGRID 65536   // 256*256 grid points

// ---------------------------------------------------------------------------
// Kernel 1: MLP 2->100->100->100->128 using V_WMMA_F32_16X16X4_F32.
// 16 blocks of exactly one wave32; block b handles batch rows [16b, 16b+16).
// ---------------------------------------------------------------------------
__global__ __launch_bounds__(32) void mlp_wmma_kernel(
    const float* __restrict__ theta,
    const float* __restrict__ W0, const float* __restrict__ b0,
    const float* __restrict__ W1, const float* __restrict__ b1,
    const float* __restrict__ W2, const float* __restrict__ b2,
    const float* __restrict__ W3, const float* __restrict__ b3,
    float* __restrict__ outbuf)   // [256][128]
{
    __shared__ float hA[16 * HID_PAD];
    __shared__ float hB[16 * HID_PAD];

    const int lane = threadIdx.x;
    const int hlf  = lane >> 4;     // 0: K={0,1} slice, 1: K={2,3} slice
    const int lm   = lane & 15;     // M row (A) / N col (B,C,D)
    const int row0 = blockIdx.x * 16;

    // ---- Layer 0: t = (theta-LO)/SCALE, A is 16x4 (K padded 2->4 w/ zeros)
    v2f a0;
    if (hlf == 0) {
        a0.x = (theta[(row0 + lm) * 2 + 0] - 50.0f)   / 40.0f;
        a0.y = (theta[(row0 + lm) * 2 + 1] - 0.0075f) / 0.0492f;
    } else {
        a0.x = 0.0f; a0.y = 0.0f;   // K=2,3 padding rows
    }
    for (int nt = 0; nt < 7; ++nt) {
        const int n  = nt * 16 + lm;
        const bool nv = (n < 100);
        v2f bm;
        if (hlf == 0 && nv) { bm.x = W0[0 * 100 + n]; bm.y = W0[1 * 100 + n]; }
        else                { bm.x = 0.0f;            bm.y = 0.0f; }
        const float bias = nv ? b0[n] : 0.0f;
        v8f acc;
        #pragma unroll
        for (int i = 0; i < 8; ++i) acc[i] = bias;
        acc = __builtin_amdgcn_wmma_f32_16x16x4_f32(false, a0, false, bm,
                                                    (short)0, acc, false, false);
        #pragma unroll
        for (int i = 0; i < 8; ++i)
            hA[(i + hlf * 8) * HID_PAD + n] = fmaxf(acc[i], 0.0f);
    }
    __syncthreads();

    // ---- Layers 1 and 2: 100 -> 100 (K = 25 steps of 4)
    for (int L = 0; L < 2; ++L) {
        const float* W  = (L == 0) ? W1 : W2;
        const float* bb = (L == 0) ? b1 : b2;
        const float* src = (L == 0) ? hA : hB;
        float*       dst = (L == 0) ? hB : hA;
        for (int nt = 0; nt < 7; ++nt) {
            const int n  = nt * 16 + lm;
            const bool nv = (n < 100);
            const float bias = nv ? bb[n] : 0.0f;
            v8f acc;
            #pragma unroll
            for (int i = 0; i < 8; ++i) acc[i] = bias;
            for (int kk = 0; kk < 25; ++kk) {
                const int k0 = kk * 4 + hlf * 2;
                v2f a;
                a.x = src[lm * HID_PAD + k0];
                a.y = src[lm * HID_PAD + k0 + 1];
                v2f bm;
                if (nv) { bm.x = W[k0 * 100 + n]; bm.y = W[(k0 + 1) * 100 + n]; }
                else    { bm.x = 0.0f;            bm.y = 0.0f; }
                acc = __builtin_amdgcn_wmma_f32_16x16x4_f32(false, a, false, bm,
                                                            (short)0, acc, false, false);
            }
            #pragma unroll
            for (int i = 0; i < 8; ++i)
                dst[(i + hlf * 8) * HID_PAD + n] = fmaxf(acc[i], 0.0f);
        }
        __syncthreads();
    }

    // ---- Layer 3: 100 -> 128, no ReLU, write to global out (row-major 256x128)
    for (int nt = 0; nt < 8; ++nt) {
        const int n = nt * 16 + lm;
        v8f acc;
        #pragma unroll
        for (int i = 0; i < 8; ++i) acc[i] = b3[n];
        for (int kk = 0; kk < 25; ++kk) {
            const int k0 = kk * 4 + hlf * 2;
            v2f a;
            a.x = hA[lm * HID_PAD + k0];
            a.y = hA[lm * HID_PAD + k0 + 1];
            v2f bm;
            bm.x = W3[k0 * 128 + n];
            bm.y = W3[(k0 + 1) * 128 + n];
            acc = __builtin_amdgcn_wmma_f32_16x16x4_f32(false, a, false, bm,
                                                        (short)0, acc, false, false);
        }
        #pragma unroll
        for (int i = 0; i < 8; ++i)
            outbuf[(size_t)(row0 + i + hlf * 8) * 128 + n] = acc[i];
    }
}

// ---------------------------------------------------------------------------
// Kernel 2: per-channel natural cubic spline fit (Thomas algorithm, n=126)
// y is outbuf viewed as [128][256] (flat reinterpretation == reshape).
// One block of 256 threads; thread j owns channel j.
// ---------------------------------------------------------------------------
__global__ __launch_bounds__(256) void spline_fit_kernel(
    const float* __restrict__ knots,   // [128]
    const float* __restrict__ y,       // [128*256]
    float* __restrict__ Mbuf,          // [128*256]
    float* __restrict__ Bc,            // [127*256]
    float* __restrict__ Cc,            // [127*256]
    float* __restrict__ Dc,            // [127*256]
    float* __restrict__ cp,            // [126*256]
    float* __restrict__ dp)            // [126*256]
{
    const int j = threadIdx.x;

    // Forward sweep. sub_k = h[k], diag_k = 2(h[k]+h[k+1]), super_k = h[k+1]
    float h0 = knots[1] - knots[0];
    float h1 = knots[2] - knots[1];
    float y0 = y[0 * NCH + j], y1 = y[1 * NCH + j], y2 = y[2 * NCH + j];
    float diag = 2.0f * (h0 + h1);
    float rhs  = 6.0f * ((y2 - y1) / h1 - (y1 - y0) / h0);
    float cprev = h1 / diag;
    float dprev = rhs / diag;
    cp[0 * NCH + j] = cprev;
    dp[0 * NCH + j] = dprev;
    for (int k = 1; k < 126; ++k) {
        const float hk  = knots[k + 1] - knots[k];
        const float hk1 = knots[k + 2] - knots[k + 1];
        const float yk  = y[k * NCH + j];
        const float yk1 = y[(k + 1) * NCH + j];
        const float yk2 = y[(k + 2) * NCH + j];
        const float dg   = 2.0f * (hk + hk1);
        const float rhsk = 6.0f * ((yk2 - yk1) / hk1 - (yk1 - yk) / hk);
        const float m  = dg - hk * cprev;
        const float ck = hk1 / m;
        const float dk = (rhsk - hk * dprev) / m;
        cp[k * NCH + j] = ck;
        dp[k * NCH + j] = dk;
        cprev = ck; dprev = dk;
    }
    // Back substitution; M[k+1] = x_k, M[0] = M[127] = 0
    float x = dp[125 * NCH + j];
    Mbuf[126 * NCH + j] = x;
    for (int k = 124; k >= 0; --k) {
        x = dp[k * NCH + j] - cp[k * NCH + j] * x;
        Mbuf[(k + 1) * NCH + j] = x;
    }
    Mbuf[0 * NCH + j]   = 0.0f;
    Mbuf[127 * NCH + j] = 0.0f;

    // Spline coefficients (a == y[:-1], stored implicitly)
    for (int k = 0; k < 127; ++k) {
        const float hk  = knots[k + 1] - knots[k];
        const float yk  = y[k * NCH + j];
        const float yk1 = y[(k + 1) * NCH + j];
        const float Mk  = Mbuf[k * NCH + j];
        const float Mk1 = Mbuf[(k + 1) * NCH + j];
        Bc[k * NCH + j] = (yk1 - yk) / hk - hk * (2.0f * Mk + Mk1) * (1.0f / 6.0f);
        Cc[k * NCH + j] = 0.5f * Mk;
        Dc[k * NCH + j] = (Mk1 - Mk) / (6.0f * hk);
    }
}

// ---------------------------------------------------------------------------
// Kernel 3: idx[g] = clip(searchsorted(knots, x, 'right')-1, 0, 126), fr[g]
// ---------------------------------------------------------------------------
__global__ __launch_bounds__(256) void searchsorted_kernel(
    const float* __restrict__ knots,
    const float* __restrict__ gridp,   // [65536]
    int* __restrict__ idxbuf,
    float* __restrict__ frbuf)
{
    const int g = blockIdx.x * 256 + threadIdx.x;
    const float x = gridp[g];
    int lo = 0, hi = NKNOT;            // count of knots <= x
    while (lo < hi) {
        const int mid = (lo + hi) >> 1;
        if (knots[mid] <= x) lo = mid + 1; else hi = mid;
    }
    int idx = lo - 1;
    idx = (idx < 0) ? 0 : ((idx > 126) ? 126 : idx);
    idxbuf[g] = idx;
    frbuf[g]  = x - knots[idx];
}

// ---------------------------------------------------------------------------
// Kernel 4: evaluation. out[ch][gi][gj] = poly(fr[g]); coalesced along g.
// ---------------------------------------------------------------------------
__global__ __launch_bounds__(256) void eval_kernel(
    const float* __restrict__ y,       // a-coeff = y[idx][ch]
    const float* __restrict__ Bc,
    const float* __restrict__ Cc,
    const float* __restrict__ Dc,
    const int* __restrict__ idxbuf,
    const float* __restrict__ frbuf,
    float* __restrict__ out)           // [256][256][256]
{
    const int g  = blockIdx.x * 256 + threadIdx.x;
    const int ch = blockIdx.y;
    const int idx = idxbuf[g];
    const float fr = frbuf[g];
    const int o = idx * NCH + ch;
    const float val = y[o] + fr * (Bc[o] + fr * (Cc[o] + fr * Dc[o]));
    out[(size_t)ch * NGRID + g] = val;
}

// ---------------------------------------------------------------------------
extern "C" void kernel_launch(void* const* d_in, const int* in_sizes, int n_in,
                              void* d_out, int out_size, void* d_ws, size_t ws_size,
                              hipStream_t stream) {
    (void)in_sizes; (void)n_in; (void)out_size; (void)ws_size;
    const float* theta = (const float*)d_in[0];
    const float* W0    = (const float*)d_in[1];
    const float* b0    = (const float*)d_in[2];
    const float* W1    = (const float*)d_in[3];
    const float* b1    = (const float*)d_in[4];
    const float* W2    = (const float*)d_in[5];
    const float* b2    = (const float*)d_in[6];
    const float* W3    = (const float*)d_in[7];
    const float* b3    = (const float*)d_in[8];
    const float* knots = (const float*)d_in[9];
    const float* gridp = (const float*)d_in[10];
    float* out = (float*)d_out;

    // workspace slabs of 32768 floats each
    float* ws     = (float*)d_ws;
    float* outbuf = ws + 0 * 32768;   // MLP output, 256x128
    float* Mbuf   = ws + 1 * 32768;   // 128x256
    float* Bc     = ws + 2 * 32768;   // 127x256
    float* Cc     = ws + 3 * 32768;
    float* Dc     = ws + 4 * 32768;
    float* cp     = ws + 5 * 32768;   // 126x256
    float* dp     = ws + 6 * 32768;
    int*   idxbuf = (int*)(ws + 7 * 32768);   // 65536 ints (2 slabs)
    float* frbuf  = ws + 9 * 32768;           // 65536 floats (2 slabs)

    mlp_wmma_kernel<<<16, 32, 0, stream>>>(theta, W0, b0, W1, b1, W2, b2, W3, b3, outbuf);
    spline_fit_kernel<<<1, 256, 0, stream>>>(knots, outbuf, Mbuf, Bc, Cc, Dc, cp, dp);
    searchsorted_kernel<<<NGRID / 256, 256, 0, stream>>>(knots, gridp, idxbuf, frbuf);
    eval_kernel<<<dim3(NGRID / 256, NCH), 256, 0, stream>>>(outbuf, Bc, Cc, Dc, idxbuf, frbuf, out);
}